// PARALLEL_SE_BLOCK_3264175145591
// MI455X (gfx1250) — compile-verified
//
#include <hip/hip_runtime.h>
#include <math.h>

typedef __attribute__((ext_vector_type(2))) float v2f;
typedef __attribute__((ext_vector_type(4))) float v4f;
typedef __attribute__((ext_vector_type(8))) float v8f;

#define Bn 16
#define Cn 256
#define Hn 128
#define Wn 128
#define CRn 16   // C / R

// ---------------------------------------------------------------------------
// Kernel 1: per-(b,c) plane, compute the 4 parity means (64x64 each).
// One block per plane (4096 blocks). Thread t handles half a row:
//   row = t>>1 (0..127), half = t&1. Row parity ro = bit1 of t.
//   float4 lanes: .x/.z are col-parity 0, .y/.w are col-parity 1.
// Tree-reduce in LDS with strides >=4 so bit1 (row parity) classes never mix.
// Branch index i = ro + 2*co  (offsets ((0,0),(1,0),(0,1),(1,1))).
// Loads are regular-temporal: we WANT fm resident in L2 for the scale pass.
// ---------------------------------------------------------------------------
__global__ __launch_bounds__(256) void se_reduce(const float* __restrict__ fm,
                                                 float* __restrict__ g) {
  __shared__ float lds_e[256];
  __shared__ float lds_o[256];
  int t = threadIdx.x;
  int bc = blockIdx.x;                               // b*256 + c
  const v4f* plane = (const v4f*)fm + (size_t)bc * (Hn * Wn / 4);
  int row  = t >> 1;
  int half = t & 1;
  int base = row * (Wn / 4) + half * (Wn / 8);       // row*32 + half*16
  float se = 0.f, so = 0.f;
#pragma unroll
  for (int j = 0; j < 16; ++j) {
    v4f v = plane[base + j];
    se += v.x + v.z;                                  // co = 0
    so += v.y + v.w;                                  // co = 1
  }
  lds_e[t] = se;
  lds_o[t] = so;
  __syncthreads();
  for (int st = 128; st >= 4; st >>= 1) {
    if (t < st) { lds_e[t] += lds_e[t + st]; lds_o[t] += lds_o[t + st]; }
    __syncthreads();
  }
  if (t == 0) {
    const float inv = 1.0f / 4096.0f;
    g[0 * 4096 + bc] = (lds_e[0] + lds_e[1]) * inv;   // ro=0, co=0 -> i=0
    g[1 * 4096 + bc] = (lds_e[2] + lds_e[3]) * inv;   // ro=1, co=0 -> i=1
    g[2 * 4096 + bc] = (lds_o[0] + lds_o[1]) * inv;   // ro=0, co=1 -> i=2
    g[3 * 4096 + bc] = (lds_o[2] + lds_o[3]) * inv;   // ro=1, co=1 -> i=3
  }
}

// ---------------------------------------------------------------------------
// Kernel 2: the 4 SE MLPs via V_WMMA_F32_16X16X4_F32. One block per branch.
//
// f32 WMMA fragment layouts (wave32):
//  A 16x4 : lanes 0-15 -> M=lane, VGPR0=K=0,VGPR1=K=1; lanes 16-31 -> K=2,3
//  B 4x16 : lanes 0-15 -> N=lane, VGPR0=K=0,VGPR1=K=1; lanes 16-31 -> K=2,3
//  C/D    : 8 VGPRs, N = lane&15, M = v + 8*(lane>>4)
//
// GEMM1 (wave 0): h = relu(g_i @ w1_i^T + b1_i)   [16x16, K=256 -> 64 WMMA]
//   A[m][k] = g_i[m*256+k], B[k][n] = w1_i[n*256+k]  (same lane expr, swapped ptr)
// GEMM2 (8 waves x 2 tiles): s = sigmoid(h @ w2_i^T + b2_i) [16x256, K=16]
// ---------------------------------------------------------------------------
__global__ __launch_bounds__(256) void se_mlp(const float* __restrict__ g,
                                              const float* __restrict__ w1,
                                              const float* __restrict__ b1,
                                              const float* __restrict__ w2,
                                              const float* __restrict__ b2,
                                              float* __restrict__ s) {
  __shared__ float hbuf[16 * 16];
  int i    = blockIdx.x;
  int t    = threadIdx.x;
  int wave = t >> 5;
  int lane = t & 31;
  const float* gi  = g  + i * 4096;
  const float* w1i = w1 + i * CRn * Cn;   // 16 x 256
  const float* b1i = b1 + i * CRn;
  const float* w2i = w2 + i * Cn * CRn;   // 256 x 16
  const float* b2i = b2 + i * Cn;

  int mn = lane & 15;                // M for A-frag, N for B-frag and D
  int hi = lane >> 4;
  int kb = hi << 1;                  // K sub-offset within the 4-wide step

  if (wave == 0) {
    v8f acc = {};
    for (int k = 0; k < Cn; k += 4) {
      v2f a, b;
      a.x = gi[mn * Cn + k + kb];
      a.y = gi[mn * Cn + k + kb + 1];
      b.x = w1i[mn * Cn + k + kb];
      b.y = w1i[mn * Cn + k + kb + 1];
      acc = __builtin_amdgcn_wmma_f32_16x16x4_f32(false, a, false, b,
                                                  (short)0, acc, false, false);
    }
    float bias = b1i[mn];
#pragma unroll
    for (int v = 0; v < 8; ++v) {
      int M = v + 8 * hi;
      float hv = acc[v] + bias;
      hbuf[M * 16 + mn] = hv > 0.f ? hv : 0.f;       // relu
    }
  }
  __syncthreads();

  // GEMM2: 16 column tiles, 2 per wave
#pragma unroll
  for (int tt = 0; tt < 2; ++tt) {
    int tile = wave * 2 + tt;
    int nb   = tile * 16;
    v8f acc = {};
#pragma unroll
    for (int k = 0; k < CRn; k += 4) {
      v2f a, b;
      a.x = hbuf[mn * 16 + k + kb];
      a.y = hbuf[mn * 16 + k + kb + 1];
      b.x = w2i[(nb + mn) * CRn + k + kb];            // B[k][n] = w2[n][k]
      b.y = w2i[(nb + mn) * CRn + k + kb + 1];
      acc = __builtin_amdgcn_wmma_f32_16x16x4_f32(false, a, false, b,
                                                  (short)0, acc, false, false);
    }
#pragma unroll
    for (int v = 0; v < 8; ++v) {
      int M = v + 8 * hi;
      int N = nb + mn;
      float x = acc[v] + b2i[N];
      s[i * 4096 + M * Cn + N] = 1.0f / (1.0f + __expf(-x));
    }
  }
}

// ---------------------------------------------------------------------------
// Kernel 3: out = fm * s[branch(b,c,parity)]. v4f-vectorized stream.
// Within a float4 (4 consecutive w): .x/.z -> co=0 (i=ro), .y/.w -> co=1 (i=ro+2)
// fm read is LAST USE -> non-temporal load (don't re-retain in L2).
// out is never re-read -> non-temporal store (don't evict remaining fm lines).
// ---------------------------------------------------------------------------
__global__ __launch_bounds__(256) void se_scale(const float* __restrict__ fm,
                                                const float* __restrict__ s,
                                                float* __restrict__ out) {
  size_t idx = (size_t)blockIdx.x * 256 + threadIdx.x;   // float4 index
  v4f v = __builtin_nontemporal_load((const v4f*)fm + idx);
  int h  = (int)((idx >> 5) & 127);                      // W/4 = 32 f4 per row
  int bc = (int)(idx >> 12);                             // 128*32 f4 per plane
  int ro = h & 1;
  float sA = s[ro * 4096 + bc];                          // i = ro   (co=0)
  float sB = s[(ro + 2) * 4096 + bc];                    // i = ro+2 (co=1)
  v4f r;
  r.x = v.x * sA;
  r.y = v.y * sB;
  r.z = v.z * sA;
  r.w = v.w * sB;
  __builtin_nontemporal_store(r, (v4f*)out + idx);
}

extern "C" void kernel_launch(void* const* d_in, const int* in_sizes, int n_in,
                              void* d_out, int out_size, void* d_ws, size_t ws_size,
                              hipStream_t stream) {
  const float* fm = (const float*)d_in[0];
  const float* w1 = (const float*)d_in[1];
  const float* b1 = (const float*)d_in[2];
  const float* w2 = (const float*)d_in[3];
  const float* b2 = (const float*)d_in[4];
  float* out = (float*)d_out;
  float* g   = (float*)d_ws;          // [4][16][256]
  float* s   = g + 4 * 4096;          // [4][16][256]

  se_reduce<<<Bn * Cn, 256, 0, stream>>>(fm, g);
  se_mlp<<<4, 256, 0, stream>>>(g, w1, b1, w2, b2, s);
  se_scale<<<(Bn * Cn * Hn * Wn / 4) / 256, 256, 0, stream>>>(fm, s, out);
}